// GCNMasker_36189394437069
// MI455X (gfx1250) — compile-verified
//
#include <hip/hip_runtime.h>
#include <math.h>

typedef float v2f __attribute__((ext_vector_type(2)));
typedef float v8f __attribute__((ext_vector_type(8)));

#define HID 64

// ---------------------------------------------------------------------------
// Degree / norm kernels: deg = 1 (self loop) + in-degree by col; dinv = rsqrt
// ---------------------------------------------------------------------------
__global__ __launch_bounds__(256) void gcn_init_deg(float* __restrict__ deg, int N) {
    int i = blockIdx.x * 256 + threadIdx.x;
    if (i < N) deg[i] = 1.0f;
}

__global__ __launch_bounds__(256) void gcn_accum_deg(float* __restrict__ deg,
                                                     const int* __restrict__ col, int E) {
    int e = blockIdx.x * 256 + threadIdx.x;
    if (e < E) atomicAdd(&deg[col[e]], 1.0f);
}

__global__ __launch_bounds__(256) void gcn_finish_dinv(float* __restrict__ dinv, int N) {
    int i = blockIdx.x * 256 + threadIdx.x;
    if (i < N) dinv[i] = rsqrtf(fmaxf(dinv[i], 1.0f));
}

// ---------------------------------------------------------------------------
// WMMA GEMM: Out[M,64] = A[M,K] @ W[K,64] (+ bias), K in {64,128}, fp32.
// One 256-thread block = 8 waves; wave w owns rows [blk*128 + w*16, +16),
// all 64 output columns as 4 accumulator tiles of 16x16.
// Uses V_WMMA_F32_16X16X4_F32; A fragment reused across the 4 column tiles.
// ---------------------------------------------------------------------------
__global__ __launch_bounds__(256) void gcn_gemm_wmma(const float* __restrict__ A,
                                                     const float* __restrict__ W,
                                                     const float* __restrict__ bias,
                                                     float* __restrict__ Out,
                                                     int M, int K) {
    __shared__ float wlds[128 * HID];  // 32 KB max (K<=128)

    for (int i = threadIdx.x; i < K * HID; i += 256) wlds[i] = W[i];
    __syncthreads();

    const int wave  = threadIdx.x >> 5;
    const int lane  = threadIdx.x & 31;
    const int lmod  = lane & 15;        // position within half-wave
    const int lhi   = (lane >> 4) & 1;  // 0: lanes 0-15, 1: lanes 16-31
    const int m0    = blockIdx.x * 128 + wave * 16;

    // A-fragment row for this lane (clamped for tail blocks; stores are guarded)
    int arow = m0 + lmod;
    if (arow > M - 1) arow = M - 1;
    const float* aptr = A + (size_t)arow * K;

    v8f acc[4] = {v8f{}, v8f{}, v8f{}, v8f{}};

    for (int k = 0; k < K; k += 4) {
        const int kk = k + (lhi ? 2 : 0);
        // A (16x4 f32): VGPR0 = K=kk, VGPR1 = K=kk+1 per ISA layout
        v2f afrag;
        afrag.x = aptr[kk];
        afrag.y = aptr[kk + 1];
#pragma unroll
        for (int t = 0; t < 4; ++t) {
            const int n = t * 16 + lmod;
            v2f bfrag;  // B (4x16 f32), symmetric layout over N
            bfrag.x = wlds[kk * HID + n];
            bfrag.y = wlds[(kk + 1) * HID + n];
            acc[t] = __builtin_amdgcn_wmma_f32_16x16x4_f32(
                false, afrag, false, bfrag, (short)0, acc[t], false, false);
        }
    }

    // C/D layout: VGPR i -> row m0+i (lanes 0-15) or m0+8+i (lanes 16-31), col n0+lmod
#pragma unroll
    for (int t = 0; t < 4; ++t) {
        const int n = t * 16 + lmod;
        const float b = bias ? bias[n] : 0.0f;
#pragma unroll
        for (int i = 0; i < 8; ++i) {
            const int m = m0 + i + lhi * 8;
            if (m < M) Out[(size_t)m * HID + n] = acc[t][i] + b;
        }
    }
}

// ---------------------------------------------------------------------------
// agg[i,:] = conv_b + m[i,:] * dinv[i]^2     (self-loop term of (A+I) norm)
// ---------------------------------------------------------------------------
__global__ __launch_bounds__(256) void gcn_init_agg(float* __restrict__ agg,
                                                    const float* __restrict__ m,
                                                    const float* __restrict__ dinv,
                                                    const float* __restrict__ bias,
                                                    int N) {
    size_t gid = (size_t)blockIdx.x * 256 + threadIdx.x;
    if (gid < (size_t)N * HID) {
        int i = (int)(gid >> 6);
        int j = (int)(gid & 63);
        float d = dinv[i];
        agg[gid] = bias[j] + m[gid] * d * d;
    }
}

// ---------------------------------------------------------------------------
// Edge scatter: agg[col,:] += m[row,:] * dinv[row]*dinv[col]
// 64 consecutive threads per edge -> coalesced reads + contiguous f32 atomics
// (targets are L2-resident: agg is 25.6 MB << 192 MB L2)
// ---------------------------------------------------------------------------
__global__ __launch_bounds__(256) void gcn_scatter(float* __restrict__ agg,
                                                   const float* __restrict__ m,
                                                   const float* __restrict__ dinv,
                                                   const int* __restrict__ row,
                                                   const int* __restrict__ col,
                                                   int E) {
    size_t gid = (size_t)blockIdx.x * 256 + threadIdx.x;
    int e = (int)(gid >> 6);
    int j = (int)(gid & 63);
    if (e < E) {
        int r = row[e];
        int c = col[e];
        float nrm = dinv[r] * dinv[c];
        atomicAdd(&agg[(size_t)c * HID + j], m[(size_t)r * HID + j] * nrm);
    }
}

__global__ __launch_bounds__(256) void gcn_relu(float* __restrict__ h,
                                                const float* __restrict__ agg,
                                                int N) {
    size_t gid = (size_t)blockIdx.x * 256 + threadIdx.x;
    if (gid < (size_t)N * HID) h[gid] = fmaxf(agg[gid], 0.0f);
}

// ---------------------------------------------------------------------------
// Per-node heads: node_score = sigmoid(h.node_w + nb); s_src/s_dst partials
// ---------------------------------------------------------------------------
__global__ __launch_bounds__(256) void gcn_node_scores(const float* __restrict__ h,
                                                       const float* __restrict__ node_w,
                                                       const float* __restrict__ node_b,
                                                       const float* __restrict__ edge_w,
                                                       float* __restrict__ node_out,
                                                       float* __restrict__ ssrc,
                                                       float* __restrict__ sdst,
                                                       int N) {
    int i = blockIdx.x * 256 + threadIdx.x;
    if (i >= N) return;
    const float* hr = h + (size_t)i * HID;
    float sn = 0.0f, sa = 0.0f, sb = 0.0f;
#pragma unroll
    for (int j = 0; j < HID; ++j) {
        float v = hr[j];
        sn += v * node_w[j];
        sa += v * edge_w[j];
        sb += v * edge_w[HID + j];
    }
    node_out[i] = 1.0f / (1.0f + __expf(-(sn + node_b[0])));
    ssrc[i] = sa;
    sdst[i] = sb;
}

__global__ __launch_bounds__(256) void gcn_edge_scores(const float* __restrict__ ssrc,
                                                       const float* __restrict__ sdst,
                                                       const int* __restrict__ row,
                                                       const int* __restrict__ col,
                                                       const float* __restrict__ edge_b,
                                                       float* __restrict__ edge_out,
                                                       int E) {
    int e = blockIdx.x * 256 + threadIdx.x;
    if (e < E) {
        float s = ssrc[row[e]] + sdst[col[e]] + edge_b[0];
        edge_out[e] = 1.0f / (1.0f + __expf(-s));
    }
}

// ---------------------------------------------------------------------------
extern "C" void kernel_launch(void* const* d_in, const int* in_sizes, int n_in,
                              void* d_out, int out_size, void* d_ws, size_t ws_size,
                              hipStream_t stream) {
    const float* x    = (const float*)d_in[0];
    const int*   ei   = (const int*)d_in[1];
    const float* fc_w = (const float*)d_in[2];
    const float* fc_b = (const float*)d_in[3];
    const float* c1w  = (const float*)d_in[4];
    const float* c1b  = (const float*)d_in[5];
    const float* c2w  = (const float*)d_in[6];
    const float* c2b  = (const float*)d_in[7];
    const float* nw   = (const float*)d_in[8];
    const float* nb   = (const float*)d_in[9];
    const float* ew   = (const float*)d_in[10];
    const float* eb   = (const float*)d_in[11];

    const int N = in_sizes[0] / 128;  // IN_DIM = 128
    const int E = in_sizes[1] / 2;
    const int* row = ei;
    const int* col = ei + E;

    // Workspace layout
    char* ws = (char*)d_ws;
    const size_t n64b = (size_t)N * HID * sizeof(float);
    float* h    = (float*)ws; ws += n64b;
    float* m    = (float*)ws; ws += n64b;
    float* agg  = (float*)ws; ws += n64b;
    float* dinv = (float*)ws; ws += (size_t)N * sizeof(float);
    float* ssrc = (float*)ws; ws += (size_t)N * sizeof(float);
    float* sdst = (float*)ws; ws += (size_t)N * sizeof(float);

    float* edge_out = (float*)d_out;
    float* node_out = (float*)d_out + E;

    const unsigned gN    = (unsigned)((N + 255) / 256);
    const unsigned gE    = (unsigned)((E + 255) / 256);
    const unsigned gN64  = (unsigned)(((size_t)N * HID + 255) / 256);
    const unsigned gE64  = (unsigned)(((size_t)E * HID + 255) / 256);
    const unsigned gGemm = (unsigned)((N + 127) / 128);

    // Degrees / symmetric normalization
    gcn_init_deg<<<gN, 256, 0, stream>>>(dinv, N);
    gcn_accum_deg<<<gE, 256, 0, stream>>>(dinv, col, E);
    gcn_finish_dinv<<<gN, 256, 0, stream>>>(dinv, N);

    // h = x @ fc_w + fc_b
    gcn_gemm_wmma<<<gGemm, 256, 0, stream>>>(x, fc_w, fc_b, h, N, 128);

    // conv1: m = h@W; agg = b + self + scatter; h = relu(agg)
    gcn_gemm_wmma<<<gGemm, 256, 0, stream>>>(h, c1w, nullptr, m, N, HID);
    gcn_init_agg<<<gN64, 256, 0, stream>>>(agg, m, dinv, c1b, N);
    gcn_scatter<<<gE64, 256, 0, stream>>>(agg, m, dinv, row, col, E);
    gcn_relu<<<gN64, 256, 0, stream>>>(h, agg, N);

    // conv2
    gcn_gemm_wmma<<<gGemm, 256, 0, stream>>>(h, c2w, nullptr, m, N, HID);
    gcn_init_agg<<<gN64, 256, 0, stream>>>(agg, m, dinv, c2b, N);
    gcn_scatter<<<gE64, 256, 0, stream>>>(agg, m, dinv, row, col, E);
    gcn_relu<<<gN64, 256, 0, stream>>>(h, agg, N);

    // Score heads
    gcn_node_scores<<<gN, 256, 0, stream>>>(h, nw, nb, ew, node_out, ssrc, sdst, N);
    gcn_edge_scores<<<gE, 256, 0, stream>>>(ssrc, sdst, row, col, eb, edge_out, E);
}